// MemoryBank_88759794139161
// MI455X (gfx1250) — compile-verified
//
#include <hip/hip_runtime.h>

typedef __attribute__((ext_vector_type(2))) float v2f;
typedef __attribute__((ext_vector_type(4))) float v4f;
typedef __attribute__((ext_vector_type(8))) float v8f;

#define B_    4
#define T_    1024
#define DK_   128
#define DV_   128
#define N_    65536
#define TOPK_ 32
#define QT    16            // queries per block (WMMA M)
#define NT    16            // keys per wave step (WMMA N)
#define NWAVES 8
#define NQ    (B_ * T_)     // 4096 queries
#define NPW   (N_ / NWAVES) // 8192 keys per wave
#define STEPS (NPW / NT)    // 512 steps

// ---------------------------------------------------------------------------
// Kernel 1: fused scores-GEMM (fp32 WMMA) + streaming exact top-32 per query
// ---------------------------------------------------------------------------
__global__ __launch_bounds__(256) void topk_score_kernel(
    const float* __restrict__ q,
    const float* __restrict__ keys,
    int* __restrict__ topk_idx)
{
    __shared__ float qs[QT][DK_];                 //  8 KB: query tile
    __shared__ float sc[NWAVES][QT][NT];          //  8 KB: per-wave score tiles
    __shared__ float topv[NWAVES][QT][TOPK_];     // 16 KB: per-wave top-32 vals
    __shared__ int   topi[NWAVES][QT][TOPK_];     // 16 KB: per-wave top-32 idx

    const int tid  = threadIdx.x;
    const int wave = tid >> 5;
    const int lane = tid & 31;
    const int gq0  = blockIdx.x * QT;             // first global query of tile

    // cooperative load of the 16x128 query tile
    for (int i = tid; i < QT * DK_; i += 256) {
        qs[i / DK_][i % DK_] = q[(size_t)gq0 * DK_ + i];
    }
    // init candidate lists
    for (int i = tid; i < NWAVES * QT * TOPK_; i += 256) {
        ((float*)topv)[i] = -3.4e38f;
        ((int*)topi)[i]   = 0;
    }
    __syncthreads();

    // shared per-lane fragment indexing for 32-bit A(16x4) and B(4x16):
    //   lanes 0-15 hold K pair {0,1}, lanes 16-31 hold K pair {2,3}
    const int row   = lane & 15;
    const int khalf = (lane >> 4) * 2;
    const int waveBase = wave * NPW;

    float minval = -3.4e38f;                      // lane-private (lanes 0-15 use it)
    int   minslot = 0;

    for (int step = 0; step < STEPS; ++step) {
        const int n0 = waveBase + step * NT;
        const float* kptr = keys + (size_t)(n0 + row) * DK_ + khalf;
        const float* qptr = &qs[row][khalf];

        // prefetch next key tile (global_prefetch_b8)
        if (step + 1 < STEPS) {
            __builtin_prefetch(kptr + (size_t)NT * DK_, 0, 0);
        }

        // 16x16 fp32 score tile: accumulate 32 x V_WMMA_F32_16X16X4_F32 over DK=128
        v8f acc = {};
        #pragma unroll
        for (int kk = 0; kk < DK_ / 4; ++kk) {
            v2f a = *(const v2f*)(qptr + kk * 4);
            v2f b = *(const v2f*)(kptr + kk * 4);
            acc = __builtin_amdgcn_wmma_f32_16x16x4_f32(
                false, a, false, b, (short)0, acc, false, false);
        }

        // spill D tile to LDS: VGPR v -> M = v + 8*(lane>>4), N = lane&15
        #pragma unroll
        for (int vv = 0; vv < 8; ++vv) {
            sc[wave][vv + 8 * (lane >> 4)][row] = acc[vv];
        }
        __syncthreads();

        // lanes 0-15: each owns one query row, exact running top-32 insert
        if (lane < 16) {
            const int qr = lane;
            #pragma unroll 1
            for (int c = 0; c < NT; ++c) {
                float sv = sc[wave][qr][c];
                if (sv > minval) {
                    topv[wave][qr][minslot] = sv;
                    topi[wave][qr][minslot] = n0 + c;
                    float mv = topv[wave][qr][0];
                    int   ms = 0;
                    #pragma unroll 1
                    for (int j = 1; j < TOPK_; ++j) {
                        float tv = topv[wave][qr][j];
                        if (tv < mv) { mv = tv; ms = j; }
                    }
                    minval = mv;
                    minslot = ms;
                }
            }
        }
        __syncthreads();
    }

    // merge: 8 waves x 32 = 256 candidates per query -> exact top-32 by rank
    const int qm  = tid >> 4;    // query row 0..15
    const int sub = tid & 15;    // 16 threads per query
    #pragma unroll 1
    for (int j = 0; j < 16; ++j) {
        const int c = sub * 16 + j;              // candidate 0..255
        const float vc = topv[c >> 5][qm][c & 31];
        const int   nc = topi[c >> 5][qm][c & 31];
        int rank = 0;
        #pragma unroll 1
        for (int k = 0; k < NWAVES * TOPK_; ++k) {
            float vk = topv[k >> 5][qm][k & 31];
            rank += (vk > vc) || (vk == vc && k < c);
        }
        if (rank < TOPK_) {
            topk_idx[(size_t)(gq0 + qm) * TOPK_ + rank] = nc;
        }
    }
}

// ---------------------------------------------------------------------------
// Kernel 2: gather K_mem / V_mem rows (dominant 128 MB store stream, float4)
// ---------------------------------------------------------------------------
__global__ __launch_bounds__(256) void gather_kernel(
    const float* __restrict__ keys,
    const float* __restrict__ values,
    const int* __restrict__ topk_idx,
    float* __restrict__ out)
{
    __shared__ int idx[TOPK_];
    const int gq  = blockIdx.x;
    const int tid = threadIdx.x;
    if (tid < TOPK_) idx[tid] = topk_idx[(size_t)gq * TOPK_ + tid];
    __syncthreads();

    const v4f* k4 = (const v4f*)keys;
    const v4f* v4 = (const v4f*)values;
    v4f* o4 = (v4f*)out;
    const size_t kout = (size_t)gq * TOPK_ * (DK_ / 4);
    const size_t vout = (size_t)NQ * TOPK_ * (DK_ / 4) + kout;

    #pragma unroll
    for (int t = 0; t < 4; ++t) {
        const int f = t * 256 + tid;             // 0..1023 float4 slots
        const int k = f >> 5;                    // selected row 0..31
        const int d = f & 31;                    // float4 within row
        const size_t src = (size_t)idx[k] * (DK_ / 4) + d;
        o4[kout + f] = k4[src];
        o4[vout + f] = v4[src];
    }
}

// ---------------------------------------------------------------------------
extern "C" void kernel_launch(void* const* d_in, const int* in_sizes, int n_in,
                              void* d_out, int out_size, void* d_ws, size_t ws_size,
                              hipStream_t stream) {
    const float* q      = (const float*)d_in[0];
    const float* keys   = (const float*)d_in[1];
    const float* values = (const float*)d_in[2];
    // d_in[3] is topk scalar (== 32, baked in at compile time)
    int* topk_idx = (int*)d_ws;                  // 4096*32 ints = 512 KB scratch

    topk_score_kernel<<<NQ / QT, 256, 0, stream>>>(q, keys, topk_idx);
    gather_kernel<<<NQ, 256, 0, stream>>>(keys, values, topk_idx, (float*)d_out);
}